// NNCLR_21002390077542
// MI455X (gfx1250) — compile-verified
//
#include <hip/hip_runtime.h>
#include <hip/hip_bf16.h>
#include <math.h>

typedef float v2f __attribute__((ext_vector_type(2)));
typedef float v8f __attribute__((ext_vector_type(8)));

#define BB   512
#define QQ   65536
#define WW   256
#define ROWS 1024          // p1 (512) + p2 (512)
#define QSLICES 64         // 1024 queue rows per slice
#define ROWTILES 64        // 1024 / 16

// ---------------------------------------------------------------------------
// Kernel 1: l2-normalize p1 and p2 into contiguous pn [1024][256]
// ---------------------------------------------------------------------------
__global__ __launch_bounds__(256) void normalize_kernel(
    const float* __restrict__ p1, const float* __restrict__ p2,
    float* __restrict__ pn)
{
    const int r = blockIdx.x;           // 0..1023
    const int t = threadIdx.x;          // 0..255 == W
    const float* src = (r < BB) ? &p1[(size_t)r * WW] : &p2[(size_t)(r - BB) * WW];
    float x = src[t];
    float ss = x * x;
    #pragma unroll
    for (int m = 16; m >= 1; m >>= 1) ss += __shfl_xor(ss, m, 32);
    __shared__ float red[8];
    if ((t & 31) == 0) red[t >> 5] = ss;
    __syncthreads();
    float tot = 0.f;
    #pragma unroll
    for (int w = 0; w < 8; ++w) tot += red[w];
    pn[(size_t)r * WW + t] = x / sqrtf(tot);
}

// ---------------------------------------------------------------------------
// Kernel 2: argmax over queue via f32 WMMA 16x16x4
// grid: (ROWTILES, QSLICES), block: 128 (4 waves). Each wave walks q-tiles
// t = wave, wave+4, ... within its 1024-row queue slice.
// ---------------------------------------------------------------------------
__global__ __launch_bounds__(128) void nn_argmax_kernel(
    const float* __restrict__ pn,      // [1024][256] normalized
    const float* __restrict__ queue,   // [65536][256] normalized
    float* __restrict__ pVal,          // [ROWTILES*QSLICES*16]
    int*   __restrict__ pIdx)          // [ROWTILES*QSLICES*16]
{
    __shared__ __align__(16) float As[16 * 260];   // padded stride: bank-conflict free
    __shared__ float wVal[4][16];
    __shared__ int   wIdx[4][16];

    const int rowBase   = blockIdx.x * 16;
    const int sliceBase = blockIdx.y * 1024;
    const int tid  = threadIdx.x;
    const int wave = tid >> 5;
    const int lane = tid & 31;
    const int mrow = lane & 15;
    const int koff = (lane & 16) ? 2 : 0;

    // Stage A tile (16 x 256 f32) into LDS
    for (int i = tid; i < 1024; i += 128) {
        const int r  = i >> 4;
        const int cv = (i & 15) << 2;
        const float4 v = *(const float4*)&pn[(size_t)(rowBase + r) * WW + cv];
        *(float4*)&As[r * 260 + cv] = v;
    }
    __syncthreads();

    float best[8];
    int   bestq[8];
    #pragma unroll
    for (int v = 0; v < 8; ++v) { best[v] = -3.0e38f; bestq[v] = 0; }

    const float* Arow = &As[mrow * 260];

    for (int t = wave; t < 64; t += 4) {
        const int qBase = sliceBase + t * 16;
        const float* Brow = &queue[(size_t)(qBase + mrow) * WW];
        v8f c = {0.f, 0.f, 0.f, 0.f, 0.f, 0.f, 0.f, 0.f};
        #pragma unroll 8
        for (int k0 = 0; k0 < WW; k0 += 4) {
            v2f a = *(const v2f*)&Arow[k0 + koff];   // LDS read
            v2f b = *(const v2f*)&Brow[k0 + koff];   // global (L2-resident)
            c = __builtin_amdgcn_wmma_f32_16x16x4_f32(
                    false, a, false, b, (short)0, c, false, false);
        }
        const int q = qBase + mrow;                  // this lane's queue column
        #pragma unroll
        for (int v = 0; v < 8; ++v) {
            if (c[v] > best[v]) { best[v] = c[v]; bestq[v] = q; }
        }
    }

    // Cross-lane argmax within each 16-lane half (masks never cross bit 4).
    #pragma unroll
    for (int v = 0; v < 8; ++v) {
        float bv = best[v]; int bq = bestq[v];
        #pragma unroll
        for (int m = 8; m >= 1; m >>= 1) {
            float ov = __shfl_xor(bv, m, 32);
            int   oq = __shfl_xor(bq, m, 32);
            if (ov > bv || (ov == bv && oq < bq)) { bv = ov; bq = oq; }
        }
        best[v] = bv; bestq[v] = bq;
    }
    if (mrow == 0) {
        const int rbase = (lane & 16) ? 8 : 0;
        #pragma unroll
        for (int v = 0; v < 8; ++v) {
            wVal[wave][rbase + v] = best[v];
            wIdx[wave][rbase + v] = bestq[v];
        }
    }
    __syncthreads();

    if (tid < 16) {
        float bv = wVal[0][tid]; int bq = wIdx[0][tid];
        #pragma unroll
        for (int w = 1; w < 4; ++w) {
            float ov = wVal[w][tid]; int oq = wIdx[w][tid];
            if (ov > bv || (ov == bv && oq < bq)) { bv = ov; bq = oq; }
        }
        const size_t o = ((size_t)blockIdx.x * gridDim.y + blockIdx.y) * 16 + tid;
        pVal[o] = bv;
        pIdx[o] = bq;
    }
}

// ---------------------------------------------------------------------------
// Kernel 3: merge slice partials -> nnIdx[1024]
// ---------------------------------------------------------------------------
__global__ __launch_bounds__(64) void nn_reduce_kernel(
    const float* __restrict__ pVal, const int* __restrict__ pIdx,
    int* __restrict__ nnIdx)
{
    const int R = blockIdx.x;          // 0..1023
    const int rowTile = R >> 4, rin = R & 15;
    const int s = threadIdx.x;         // slice 0..63
    __shared__ float sv[64];
    __shared__ int   sq[64];
    sv[s] = pVal[((size_t)rowTile * QSLICES + s) * 16 + rin];
    sq[s] = pIdx[((size_t)rowTile * QSLICES + s) * 16 + rin];
    __syncthreads();
    if (s == 0) {
        float bv = sv[0]; int bq = sq[0];
        for (int i = 1; i < 64; ++i) {
            if (sv[i] > bv || (sv[i] == bv && sq[i] < bq)) { bv = sv[i]; bq = sq[i]; }
        }
        nnIdx[R] = bq;
    }
}

// ---------------------------------------------------------------------------
// Kernel 4: S[z] = queue[nn_z] @ pOther.T * (1/T)   (two 512x512 GEMMs, WMMA)
// z=0: nn1 vs p2n ; z=1: nn2 vs p1n. One wave per 16x16 output tile.
// ---------------------------------------------------------------------------
__global__ __launch_bounds__(32) void logits_gemm_kernel(
    const float* __restrict__ pn, const float* __restrict__ queue,
    const int* __restrict__ nnIdx, float* __restrict__ S)
{
    const int z = blockIdx.z;
    const int rowBase = blockIdx.x * 16;
    const int colBase = blockIdx.y * 16;
    const int lane = threadIdx.x & 31;
    const int mrow = lane & 15;
    const int koff = (lane & 16) ? 2 : 0;

    const int aRow = nnIdx[z * BB + rowBase + mrow];
    const float* Arow = &queue[(size_t)aRow * WW];
    // z==0 -> B from p2n (pn rows 512..1023); z==1 -> B from p1n (rows 0..511)
    const float* Brow = &pn[(size_t)(((z == 0) ? BB : 0) + colBase + mrow) * WW];

    v8f c = {0.f, 0.f, 0.f, 0.f, 0.f, 0.f, 0.f, 0.f};
    #pragma unroll 8
    for (int k0 = 0; k0 < WW; k0 += 4) {
        v2f a = *(const v2f*)&Arow[k0 + koff];
        v2f b = *(const v2f*)&Brow[k0 + koff];
        c = __builtin_amdgcn_wmma_f32_16x16x4_f32(
                false, a, false, b, (short)0, c, false, false);
    }
    float* Sz = S + (size_t)z * BB * BB;
    const int half = (lane & 16) ? 8 : 0;
    #pragma unroll
    for (int v = 0; v < 8; ++v) {
        Sz[(size_t)(rowBase + half + v) * BB + colBase + mrow] = c[v] * 10.0f; // 1/T
    }
}

// ---------------------------------------------------------------------------
// Kernel 5: per-row log-softmax loss over the virtual [2048,512] logits.
// kind 0: S1 row rr; kind 1: S1 col rr; kind 2: S2 row rr; kind 3: S2 col rr.
// Target logit is the diagonal S[rr][rr] in every mode.
// ---------------------------------------------------------------------------
__global__ __launch_bounds__(256) void loss_kernel(
    const float* __restrict__ S, float* __restrict__ out)
{
    const int r    = blockIdx.x;          // 0..2047
    const int kind = r >> 9;
    const int rr   = r & (BB - 1);
    const float* M = S + ((kind >= 2) ? (size_t)BB * BB : 0);
    const bool colMode = (kind & 1);
    const int tid = threadIdx.x;
    __shared__ float red[8];

    float vals[2];
    float mx = -3.0e38f;
    #pragma unroll
    for (int i = 0; i < 2; ++i) {
        const int j = tid + i * 256;
        const float v = colMode ? M[(size_t)j * BB + rr] : M[(size_t)rr * BB + j];
        vals[i] = v;
        mx = fmaxf(mx, v);
    }
    #pragma unroll
    for (int m = 16; m >= 1; m >>= 1) mx = fmaxf(mx, __shfl_xor(mx, m, 32));
    if ((tid & 31) == 0) red[tid >> 5] = mx;
    __syncthreads();
    float bm = red[0];
    #pragma unroll
    for (int w = 1; w < 8; ++w) bm = fmaxf(bm, red[w]);
    __syncthreads();

    float s = expf(vals[0] - bm) + expf(vals[1] - bm);
    #pragma unroll
    for (int m = 16; m >= 1; m >>= 1) s += __shfl_xor(s, m, 32);
    if ((tid & 31) == 0) red[tid >> 5] = s;
    __syncthreads();
    if (tid == 0) {
        float tot = 0.f;
        #pragma unroll
        for (int w = 0; w < 8; ++w) tot += red[w];
        const float tgt = M[(size_t)rr * BB + rr];
        out[r] = logf(tot) + bm - tgt;    // -(tgt - logsumexp)
    }
}

// ---------------------------------------------------------------------------
extern "C" void kernel_launch(void* const* d_in, const int* in_sizes, int n_in,
                              void* d_out, int out_size, void* d_ws, size_t ws_size,
                              hipStream_t stream) {
    const float* p1    = (const float*)d_in[0];   // [512,256]
    const float* p2    = (const float*)d_in[1];   // [512,256]
    const float* queue = (const float*)d_in[2];   // [65536,256]
    float* out = (float*)d_out;                   // [2048]

    char* ws = (char*)d_ws;
    float* pn    = (float*)ws;                                   // 1,048,576 B
    float* pVal  = (float*)(ws + 1048576);                       //   262,144 B
    int*   pIdx  = (int*)  (ws + 1048576 + 262144);              //   262,144 B
    int*   nnIdx = (int*)  (ws + 1048576 + 524288);              //     4,096 B
    float* S     = (float*)(ws + 1048576 + 524288 + 4096);       // 2,097,152 B

    normalize_kernel  <<<ROWS, 256, 0, stream>>>(p1, p2, pn);
    nn_argmax_kernel  <<<dim3(ROWTILES, QSLICES), 128, 0, stream>>>(pn, queue, pVal, pIdx);
    nn_reduce_kernel  <<<ROWS, 64, 0, stream>>>(pVal, pIdx, nnIdx);
    logits_gemm_kernel<<<dim3(32, 32, 2), 32, 0, stream>>>(pn, queue, nnIdx, S);
    loss_kernel       <<<4 * BB, 256, 0, stream>>>(S, out);
}